// GIN_68702296866877
// MI455X (gfx1250) — compile-verified
//
#include <hip/hip_runtime.h>

typedef __attribute__((ext_vector_type(2))) float v2f;
typedef __attribute__((ext_vector_type(8))) float v8f;

#define ROI      268
#define NGRAPH   512
#define CH       32
#define SLOPE    0.33f
#define BN_EPS   1e-5f
#define LDSTRIDE 33   // padded row stride (floats) to avoid 32-stride bank conflicts

__device__ __forceinline__ float leaky(float h) { return h > 0.0f ? h : SLOPE * h; }

// ---------------------------------------------------------------------------
// Generic GEMM: C[rows x 32] = A[rows x K] @ B[K x 32]   (rows % 16 == 0)
// One wave computes a 16-row stripe (two 16x16 N-tiles) with
// v_wmma_f32_16x16x4_f32, K-loop in steps of 4.
// A layout (ISA 7.12.2, 32-bit A 16x4): lane m = lane&15, k = v + 2*(lane>>4)
// C/D layout: VGPR r -> row = r + 8*(lane>>4), col = lane&15
// ---------------------------------------------------------------------------
__global__ void gemm_n32(const float* __restrict__ A, const float* __restrict__ B,
                         float* __restrict__ C, int K, int nRows) {
  int wave = threadIdx.x >> 5;
  int lane = threadIdx.x & 31;
  int row0 = (blockIdx.x * 4 + wave) * 16;
  if (row0 >= nRows) return;
  int m    = lane & 15;
  int half = lane >> 4;

  const float* arow = A + (size_t)(row0 + m) * K;
  v8f c0 = {};  // cols 0..15
  v8f c1 = {};  // cols 16..31

  for (int k0 = 0; k0 < K; k0 += 4) {
    int ka = k0 + 2 * half;
    v2f a, b0, b1;
    a.x  = arow[ka];
    a.y  = arow[ka + 1];
    b0.x = B[(size_t)ka * 32 + m];
    b0.y = B[(size_t)(ka + 1) * 32 + m];
    b1.x = B[(size_t)ka * 32 + 16 + m];
    b1.y = B[(size_t)(ka + 1) * 32 + 16 + m];
    c0 = __builtin_amdgcn_wmma_f32_16x16x4_f32(false, a, false, b0, (short)0, c0, false, false);
    c1 = __builtin_amdgcn_wmma_f32_16x16x4_f32(false, a, false, b1, (short)0, c1, false, false);
  }

  for (int r = 0; r < 8; ++r) {
    size_t row = (size_t)(row0 + r + 8 * half);
    C[row * 32 + m]      = c0[r];
    C[row * 32 + 16 + m] = c1[r];
  }
}

// ---------------------------------------------------------------------------
// Fused: u = leaky((1+eps)*y + agg + b1);  h = leaky(BN(u @ W2 + b2))
// Same wave/tile structure as gemm_n32, K = 32 fixed.
// ---------------------------------------------------------------------------
__global__ void gin_mlp2(const float* __restrict__ y, const float* __restrict__ agg,
                         const float* __restrict__ epsP, const float* __restrict__ b1,
                         const float* __restrict__ W2, const float* __restrict__ b2,
                         const float* __restrict__ bng, const float* __restrict__ bnb,
                         const float* __restrict__ bnm, const float* __restrict__ bnv,
                         float* __restrict__ out, int nRows) {
  int wave = threadIdx.x >> 5;
  int lane = threadIdx.x & 31;
  int row0 = (blockIdx.x * 4 + wave) * 16;
  if (row0 >= nRows) return;
  int m    = lane & 15;
  int half = lane >> 4;
  float ep = 1.0f + epsP[0];

  size_t arow = (size_t)(row0 + m) * 32;
  v8f c0 = {};
  v8f c1 = {};

  for (int k0 = 0; k0 < 32; k0 += 4) {
    int ka = k0 + 2 * half;
    v2f a, b0, b1v;
    a.x  = leaky(ep * y[arow + ka]     + agg[arow + ka]     + b1[ka]);
    a.y  = leaky(ep * y[arow + ka + 1] + agg[arow + ka + 1] + b1[ka + 1]);
    b0.x = W2[(size_t)ka * 32 + m];
    b0.y = W2[(size_t)(ka + 1) * 32 + m];
    b1v.x = W2[(size_t)ka * 32 + 16 + m];
    b1v.y = W2[(size_t)(ka + 1) * 32 + 16 + m];
    c0 = __builtin_amdgcn_wmma_f32_16x16x4_f32(false, a, false, b0,  (short)0, c0, false, false);
    c1 = __builtin_amdgcn_wmma_f32_16x16x4_f32(false, a, false, b1v, (short)0, c1, false, false);
  }

  // epilogue: bias + BN + leaky
  float s0 = rsqrtf(bnv[m] + BN_EPS)      * bng[m];
  float s1 = rsqrtf(bnv[16 + m] + BN_EPS) * bng[16 + m];
  for (int r = 0; r < 8; ++r) {
    size_t row = (size_t)(row0 + r + 8 * half);
    float v0 = (c0[r] + b2[m]      - bnm[m])      * s0 + bnb[m];
    float v1 = (c1[r] + b2[16 + m] - bnm[16 + m]) * s1 + bnb[16 + m];
    out[row * 32 + m]      = leaky(v0);
    out[row * 32 + 16 + m] = leaky(v1);
  }
}

// ---------------------------------------------------------------------------
// Edge binning (done once, reused by both GIN layers).
// gid = dst / ROI is valid because edges are constrained within a graph.
// ---------------------------------------------------------------------------
__global__ void zero_i32(int* __restrict__ p, int n) {
  int i = blockIdx.x * blockDim.x + threadIdx.x;
  if (i < n) p[i] = 0;
}

__global__ void edge_count(const int* __restrict__ dst, int* __restrict__ gcount, int E) {
  __shared__ int hist[NGRAPH];
  for (int i = threadIdx.x; i < NGRAPH; i += blockDim.x) hist[i] = 0;
  __syncthreads();
  int i = blockIdx.x * blockDim.x + threadIdx.x;
  int stride = gridDim.x * blockDim.x;
  for (; i < E; i += stride) atomicAdd(&hist[dst[i] / ROI], 1);
  __syncthreads();
  for (int j = threadIdx.x; j < NGRAPH; j += blockDim.x)
    if (hist[j]) atomicAdd(&gcount[j], hist[j]);
}

// single block of NGRAPH threads: exclusive scan -> off[0..NGRAPH], cursor=off
__global__ void scan_counts(const int* __restrict__ gcount, int* __restrict__ off,
                            int* __restrict__ cursor) {
  __shared__ int s[NGRAPH];
  int t = threadIdx.x;
  int my = gcount[t];
  s[t] = my;
  __syncthreads();
  for (int d = 1; d < NGRAPH; d <<= 1) {
    int v = (t >= d) ? s[t - d] : 0;
    __syncthreads();
    s[t] += v;
    __syncthreads();
  }
  int excl = s[t] - my;
  off[t]    = excl;
  cursor[t] = excl;
  if (t == NGRAPH - 1) off[NGRAPH] = s[t];
}

__global__ void edge_bin(const int* __restrict__ src, const int* __restrict__ dst,
                         int* __restrict__ cursor, int* __restrict__ bsrc,
                         int* __restrict__ bloc, int E) {
  int i = blockIdx.x * blockDim.x + threadIdx.x;
  if (i >= E) return;
  int d = dst[i];
  int g = d / ROI;
  int slot = atomicAdd(&cursor[g], 1);
  bsrc[slot] = src[i];
  bloc[slot] = d - g * ROI;
}

// ---------------------------------------------------------------------------
// Per-graph aggregation: agg[dst] += y[src] accumulated in LDS (ds_add_f32),
// then written out densely (no global atomics, no agg pre-clear needed).
// ---------------------------------------------------------------------------
__global__ void __launch_bounds__(256) graph_agg(const int* __restrict__ off,
                                                 const int* __restrict__ bsrc,
                                                 const int* __restrict__ bloc,
                                                 const float* __restrict__ y,
                                                 float* __restrict__ agg) {
  __shared__ float acc[ROI * LDSTRIDE];
  int g = blockIdx.x;
  for (int i = threadIdx.x; i < ROI * LDSTRIDE; i += blockDim.x) acc[i] = 0.0f;
  __syncthreads();

  int e0 = off[g], e1 = off[g + 1];
  for (int i = e0 + threadIdx.x; i < e1; i += blockDim.x) {
    int s = bsrc[i];
    int r = bloc[i];
    const float4* ys = (const float4*)(y + (size_t)s * 32);
    float* ar = acc + r * LDSTRIDE;
    for (int c4 = 0; c4 < 8; ++c4) {
      float4 v = ys[c4];
      atomicAdd(ar + c4 * 4 + 0, v.x);
      atomicAdd(ar + c4 * 4 + 1, v.y);
      atomicAdd(ar + c4 * 4 + 2, v.z);
      atomicAdd(ar + c4 * 4 + 3, v.w);
    }
  }
  __syncthreads();

  size_t base = (size_t)g * ROI * 32;
  for (int i = threadIdx.x; i < ROI * 32; i += blockDim.x) {
    int r = i >> 5, c = i & 31;
    agg[base + i] = acc[r * LDSTRIDE + c];
  }
}

// ---------------------------------------------------------------------------
// Utility / pooling / head kernels
// ---------------------------------------------------------------------------
__global__ void zero_f32(float* __restrict__ p, long long n) {
  long long i = (long long)blockIdx.x * blockDim.x + threadIdx.x;
  long long stride = (long long)gridDim.x * blockDim.x;
  for (; i < n; i += stride) p[i] = 0.0f;
}

__global__ void count_nodes(const int* __restrict__ batch, float* __restrict__ counts, int n) {
  int i = blockIdx.x * blockDim.x + threadIdx.x;
  if (i < n) atomicAdd(&counts[batch[i]], 1.0f);
}

// block per graph: max + mean over the graph's ROI contiguous rows, 32 channels
__global__ void pool_graph(const float* __restrict__ h, const float* __restrict__ counts,
                           float* __restrict__ z, int zoff) {
  int g = blockIdx.x;
  int c = threadIdx.x & 31;
  int t = threadIdx.x >> 5;  // 0..7
  float mx = -3.402823466e38f, sm = 0.0f;
  for (int r = t; r < ROI; r += 8) {
    float v = h[((size_t)g * ROI + r) * 32 + c];
    mx = fmaxf(mx, v);
    sm += v;
  }
  __shared__ float smax[8][32];
  __shared__ float ssum[8][32];
  smax[t][c] = mx;
  ssum[t][c] = sm;
  __syncthreads();
  if (t == 0) {
    float M = smax[0][c], S = ssum[0][c];
    for (int i = 1; i < 8; ++i) { M = fmaxf(M, smax[i][c]); S += ssum[i][c]; }
    z[(size_t)g * 128 + zoff + c]      = M;
    z[(size_t)g * 128 + zoff + 32 + c] = S / counts[g];
  }
}

// head layer: out = leaky(BN(in @ W + b)); one thread per (graph, out-ch)
__global__ void head_lin(const float* __restrict__ in, const float* __restrict__ W,
                         const float* __restrict__ b,
                         const float* __restrict__ g, const float* __restrict__ bb,
                         const float* __restrict__ mm, const float* __restrict__ vv,
                         float* __restrict__ out, int IN, int OUT) {
  int idx = blockIdx.x * blockDim.x + threadIdx.x;
  if (idx >= NGRAPH * OUT) return;
  int gr = idx / OUT, o = idx % OUT;
  const float* row = in + (size_t)gr * IN;
  float acc = b[o];
  for (int k = 0; k < IN; ++k) acc += row[k] * W[(size_t)k * OUT + o];
  acc = (acc - mm[o]) * rsqrtf(vv[o] + BN_EPS) * g[o] + bb[o];
  out[idx] = leaky(acc);
}

__global__ void head_final(const float* __restrict__ in, const float* __restrict__ W,
                           const float* __restrict__ b, float* __restrict__ out) {
  int gr = blockIdx.x * blockDim.x + threadIdx.x;
  if (gr >= NGRAPH) return;
  float acc = b[0];
  for (int k = 0; k < 128; ++k) acc += in[(size_t)gr * 128 + k] * W[k];
  out[gr] = acc;
}

// ---------------------------------------------------------------------------
extern "C" void kernel_launch(void* const* d_in, const int* in_sizes, int n_in,
                              void* d_out, int out_size, void* d_ws, size_t ws_size,
                              hipStream_t stream) {
  const float* x      = (const float*)d_in[0];
  const int*   eidx   = (const int*)d_in[1];
  const int*   batch  = (const int*)d_in[2];
  const float* eps0   = (const float*)d_in[3];
  const float* W1_0   = (const float*)d_in[4];
  const float* b1_0   = (const float*)d_in[5];
  const float* W2_0   = (const float*)d_in[6];
  const float* b2_0   = (const float*)d_in[7];
  const float* bn0_g  = (const float*)d_in[8];
  const float* bn0_b  = (const float*)d_in[9];
  const float* bn0_m  = (const float*)d_in[10];
  const float* bn0_v  = (const float*)d_in[11];
  const float* eps1   = (const float*)d_in[12];
  const float* W1_1   = (const float*)d_in[13];
  const float* b1_1   = (const float*)d_in[14];
  const float* W2_1   = (const float*)d_in[15];
  const float* b2_1   = (const float*)d_in[16];
  const float* bn1_g  = (const float*)d_in[17];
  const float* bn1_b  = (const float*)d_in[18];
  const float* bn1_m  = (const float*)d_in[19];
  const float* bn1_v  = (const float*)d_in[20];
  const float* lin1_W = (const float*)d_in[21];
  const float* lin1_b = (const float*)d_in[22];
  const float* hbn1_g = (const float*)d_in[23];
  const float* hbn1_b = (const float*)d_in[24];
  const float* hbn1_m = (const float*)d_in[25];
  const float* hbn1_v = (const float*)d_in[26];
  const float* lin2_W = (const float*)d_in[27];
  const float* lin2_b = (const float*)d_in[28];
  const float* hbn2_g = (const float*)d_in[29];
  const float* hbn2_b = (const float*)d_in[30];
  const float* hbn2_m = (const float*)d_in[31];
  const float* hbn2_v = (const float*)d_in[32];
  const float* lin3_W = (const float*)d_in[33];
  const float* lin3_b = (const float*)d_in[34];

  const int N = in_sizes[0] / ROI;   // 137216
  const int E = in_sizes[1] / 2;     // 1000000
  const int* src = eidx;
  const int* dst = eidx + E;

  // workspace layout
  float* ws = (float*)d_ws;
  size_t NA = (size_t)N * 32;
  float* bufA   = ws;                 // y (post-W1) buffer
  float* bufB   = bufA + NA;          // agg buffer
  float* bufC   = bufB + NA;          // h buffer
  float* z      = bufC + NA;                  // [G,128] pooled features
  float* z1     = z  + (size_t)NGRAPH * 128;  // [G,256]
  float* z2     = z1 + (size_t)NGRAPH * 256;  // [G,128]
  float* counts = z2 + (size_t)NGRAPH * 128;  // [G]
  int*   gcount = (int*)(counts + NGRAPH);    // [G]
  int*   off    = gcount + NGRAPH;            // [G+1]
  int*   cursor = off + NGRAPH + 1;           // [G]
  int*   bsrc   = cursor + NGRAPH;            // [E]
  int*   bloc   = bsrc + E;                   // [E]

  const int gemmBlocks = (N / 16 + 3) / 4;   // 4 waves (16-row tiles) per block

  // ---- counts + edge binning (reused by both layers) ----
  zero_f32<<<1, 256, 0, stream>>>(counts, NGRAPH);
  count_nodes<<<(N + 255) / 256, 256, 0, stream>>>(batch, counts, N);
  zero_i32<<<2, 256, 0, stream>>>(gcount, NGRAPH);
  edge_count<<<512, 256, 0, stream>>>(dst, gcount, E);
  scan_counts<<<1, NGRAPH, 0, stream>>>(gcount, off, cursor);
  edge_bin<<<(E + 255) / 256, 256, 0, stream>>>(src, dst, cursor, bsrc, bloc, E);

  // ---- GIN layer 0 ----
  gemm_n32<<<gemmBlocks, 128, 0, stream>>>(x, W1_0, bufA, ROI, N);      // y0 = x @ W1_0
  graph_agg<<<NGRAPH, 256, 0, stream>>>(off, bsrc, bloc, bufA, bufB);
  gin_mlp2<<<gemmBlocks, 128, 0, stream>>>(bufA, bufB, eps0, b1_0, W2_0, b2_0,
                                           bn0_g, bn0_b, bn0_m, bn0_v, bufC, N);
  pool_graph<<<NGRAPH, 256, 0, stream>>>(bufC, counts, z, 0);

  // ---- GIN layer 1 ----
  gemm_n32<<<gemmBlocks, 128, 0, stream>>>(bufC, W1_1, bufA, CH, N);    // y1 = h0 @ W1_1
  graph_agg<<<NGRAPH, 256, 0, stream>>>(off, bsrc, bloc, bufA, bufB);
  gin_mlp2<<<gemmBlocks, 128, 0, stream>>>(bufA, bufB, eps1, b1_1, W2_1, b2_1,
                                           bn1_g, bn1_b, bn1_m, bn1_v, bufC, N);
  pool_graph<<<NGRAPH, 256, 0, stream>>>(bufC, counts, z, 64);

  // ---- readout head ----
  head_lin<<<(NGRAPH * 256 + 255) / 256, 256, 0, stream>>>(z, lin1_W, lin1_b,
      hbn1_g, hbn1_b, hbn1_m, hbn1_v, z1, 128, 256);
  head_lin<<<(NGRAPH * 128 + 255) / 256, 256, 0, stream>>>(z1, lin2_W, lin2_b,
      hbn2_g, hbn2_b, hbn2_m, hbn2_v, z2, 256, 128);
  head_final<<<(NGRAPH + 255) / 256, 256, 0, stream>>>(z2, lin3_W, lin3_b, (float*)d_out);
}